// BarycentricCoordinates_82016695485070
// MI455X (gfx1250) — compile-verified
//
#include <hip/hip_runtime.h>
#include <math.h>

// CDNA5 / gfx1250: wave32, WMMA 16x16x4 f32.
typedef __attribute__((ext_vector_type(2))) float v2f;
typedef __attribute__((ext_vector_type(8))) float v8f;

#define WAVES_PER_BLOCK 8
#define BLOCK_SIZE 256
#define NPTS 16              // projections per view (N)
#define MPAIR 15             // N-1

// Wave-synchronous LDS fence: lanes are lockstep and per-wave DS ops execute
// in order, so we only need to stop the compiler from reordering LDS accesses.
#define WAVE_SYNC() __builtin_amdgcn_wave_barrier()

__global__ __launch_bounds__(BLOCK_SIZE)
void BarycentricCoordinates_82016695485070_kernel(
    const float* __restrict__ tmpl,   // (RA, 2) flattened (R*A = RA)
    const float* __restrict__ proj,   // (V, 16, 2)
    float* __restrict__ wout,         // (V, RA, 3) float32 weights
    int*   __restrict__ iout,         // (V, RA, 3) int32 indices
    int itersPerWave,                 // RA / WAVES_PER_BLOCK (uniform)
    int RA)
{
    __shared__ float proj_s[NPTS * 2];                    // this view's 16 points
    __shared__ float sDist[WAVES_PER_BLOCK][NPTS];        // sorted distances
    __shared__ int   sOrder[WAVES_PER_BLOCK][NPTS];       // argsort order
    __shared__ float sG[WAVES_PER_BLOCK][16 * 16];        // Gram matrix per wave

    const int lane = threadIdx.x & 31;
    const int wid  = threadIdx.x >> 5;
    const int v    = blockIdx.x;

    if (threadIdx.x < NPTS * 2)
        proj_s[threadIdx.x] = proj[(size_t)v * (NPTS * 2) + threadIdx.x];
    __syncthreads();   // proj_s shared across waves: real barrier, once

    for (int it = 0; it < itersPerWave; ++it) {
        const int item = it * WAVES_PER_BLOCK + wid;      // flat (r,a) index
        const float tx = tmpl[item * 2 + 0];              // wave-uniform scalar loads
        const float ty = tmpl[item * 2 + 1];

        // ---- distances: lane i (i<16) handles projection point i ----
        float d = 3.0e38f;
        if (lane < NPTS) {
            float dx = tx - proj_s[lane * 2 + 0];
            float dy = ty - proj_s[lane * 2 + 1];
            d = sqrtf(dx * dx + dy * dy);
        }

        // ---- stable ascending rank via 16 shuffles (== jnp.argsort stable) ----
        int rank = 0;
        #pragma unroll
        for (int j = 0; j < NPTS; ++j) {
            float dj = __shfl(d, j, 32);
            rank += (dj < d) || (dj == d && j < lane);
        }
        if (lane < NPTS) { sOrder[wid][rank] = lane; sDist[wid][rank] = d; }
        WAVE_SYNC();

        const int   ci = sOrder[wid][0];                  // p_closest_idx
        const float cd = sDist[wid][0];                   // closest_d
        const float pcx = proj_s[ci * 2 + 0];
        const float pcy = proj_s[ci * 2 + 1];

        // ---- A fragment: rows 0..14 = v0v1[n], row 15 = v2; K padded to 4 ----
        // 16x4 f32 A layout: lanes 0-15 hold (K=0,K=1); lanes 16-31 hold (K=2,K=3)=0.
        v2f afrag = {0.0f, 0.0f};
        if (lane < MPAIR) {
            int oi = sOrder[wid][lane + 1];
            afrag.x = proj_s[oi * 2 + 0] - pcx;
            afrag.y = proj_s[oi * 2 + 1] - pcy;
        } else if (lane == MPAIR) {
            afrag.x = tx - pcx;
            afrag.y = ty - pcy;
        }

        // ---- G = A * A^T in ONE v_wmma_f32_16x16x4_f32 ----
        // B (4x16) lane layout mirrors A, so reusing afrag gives the transpose.
        v8f g = {0.f, 0.f, 0.f, 0.f, 0.f, 0.f, 0.f, 0.f};
        g = __builtin_amdgcn_wmma_f32_16x16x4_f32(
                /*neg_a=*/false, afrag, /*neg_b=*/false, afrag,
                /*c_mod=*/(short)0, g, /*reuse_a=*/false, /*reuse_b=*/false);

        // ---- spill G to LDS (C/D layout: VGPR j, lane L -> row j+8*(L>=16), col L&15) ----
        {
            const int col = lane & 15;
            const int rowBase = (lane >> 4) << 3;
            #pragma unroll
            for (int j = 0; j < 8; ++j)
                sG[wid][(rowBase + j) * 16 + col] = g[j];
        }
        WAVE_SYNC();

        // ---- scan 225 (n,m) pairs, division-free ----
        // bc_k = N_k/denom in [0,1]  <=>  min(0,denom) <= N_k <= max(0,denom)
        const float INF = __builtin_inff();
        float bestScore = INF;
        int   bestP = 0x7fffffff;
        int n = lane / MPAIR;             // one magic-number divide, then incremental
        int m = lane - n * MPAIR;
        #pragma unroll
        for (int s = 0; s < 8; ++s) {
            const int p = lane + 32 * s;
            if (p < MPAIR * MPAIR) {
                float Gnn = sG[wid][n * 16 + n];      // dot00[n]
                float Gmm = sG[wid][m * 16 + m];      // dot00[m]
                float Gnm = sG[wid][n * 16 + m];      // dot01[n][m]
                float Dn  = sG[wid][n * 16 + 15];     // dot02[n]
                float Dm  = sG[wid][m * 16 + 15];     // dot02[m]
                float denom = Gnn * Gmm - Gnm * Gnm;
                float N2 = Dn * Gmm - Gnm * Dm;       // p2 * denom
                float N1 = Gnn * Dm - Gnm * Dn;       // p1 * denom
                float N0 = denom - N2 - N1;           // p0 * denom
                float lo = fminf(denom, 0.0f);
                float hi = fmaxf(denom, 0.0f);
                bool bad = (denom == 0.0f) || (n == m)
                        || !(N2 >= lo && N2 <= hi)
                        || !(N1 >= lo && N1 <= hi)
                        || !(N0 >= lo && N0 <= hi);
                float total = sDist[wid][n + 1] + sDist[wid][m + 1] + cd;
                float score = bad ? INF : total;
                bool better = (score < bestScore) || (score == bestScore && p < bestP);
                bestScore = better ? score : bestScore;
                bestP     = better ? p : bestP;
            }
            // advance flat index by 32: +2 rows, +2 cols, carry at 15
            n += 2; m += 2;
            if (m >= MPAIR) { m -= MPAIR; n += 1; }
        }
        // wave32 min-reduction (score, then smallest flat index)
        #pragma unroll
        for (int off = 16; off > 0; off >>= 1) {
            float s2 = __shfl_xor(bestScore, off, 32);
            int   q2 = __shfl_xor(bestP, off, 32);
            if (s2 < bestScore || (s2 == bestScore && q2 < bestP)) {
                bestScore = s2; bestP = q2;
            }
        }

        // ---- lane 0: exact reference formulas (one division) at the winner ----
        if (lane == 0) {
            int nn = bestP / MPAIR, mm = bestP % MPAIR;
            float Gnn = sG[wid][nn * 16 + nn];
            float Gmm = sG[wid][mm * 16 + mm];
            float Gnm = sG[wid][nn * 16 + mm];
            float Dn  = sG[wid][nn * 16 + 15];
            float Dm  = sG[wid][mm * 16 + 15];
            float denom = Gnn * Gmm - Gnm * Gnm;
            bool  zero  = (denom == 0.0f) || (nn == mm);
            float dinv  = zero ? 0.0f : 1.0f / denom;
            float b2 = (Dn * Gmm - Gnm * Dm) * dinv;
            float b1 = (Gnn * Dm - Gnm * Dn) * dinv;
            float b0 = 1.0f - b2 - b1;
            bool invalid = zero
                || __builtin_isnan(b0) || __builtin_isinf(b0)
                || __builtin_isnan(b1) || __builtin_isinf(b1)
                || __builtin_isnan(b2) || __builtin_isinf(b2);
            size_t base = ((size_t)v * RA + item) * 3;
            // reference order: bc = [p0, p2, p1]
            wout[base + 0] = invalid ? 0.0f : b0;
            wout[base + 1] = invalid ? 0.0f : b2;
            wout[base + 2] = invalid ? 0.0f : b1;
            iout[base + 0] = invalid ? 0 : ci;
            iout[base + 1] = invalid ? 0 : sOrder[wid][nn + 1];
            iout[base + 2] = invalid ? 0 : sOrder[wid][mm + 1];
        }
        WAVE_SYNC();   // protect per-wave LDS reuse across iterations
    }
}

extern "C" void kernel_launch(void* const* d_in, const int* in_sizes, int n_in,
                              void* d_out, int out_size, void* d_ws, size_t ws_size,
                              hipStream_t stream) {
    (void)n_in; (void)d_ws; (void)ws_size; (void)out_size;
    const float* tmpl = (const float*)d_in[0];     // (R, A, 2)
    const float* proj = (const float*)d_in[1];     // (V, 16, 2)
    const int RA = in_sizes[0] / 2;                // R*A = 40
    const int V  = in_sizes[1] / (NPTS * 2);       // 2048
    // Outputs concatenated flat: weights f32 (V*RA*3), then indices i32 (V*RA*3).
    float* wout = (float*)d_out;
    int*   iout = (int*)((float*)d_out + (size_t)V * RA * 3);
    const int itersPerWave = RA / WAVES_PER_BLOCK; // 40/8 = 5, uniform
    BarycentricCoordinates_82016695485070_kernel<<<V, BLOCK_SIZE, 0, stream>>>(
        tmpl, proj, wout, iout, itersPerWave, RA);
}